// ChannelDisassembly_6270652252218
// MI455X (gfx1250) — compile-verified
//
#include <hip/hip_runtime.h>
#include <stdint.h>

// ---------------- problem constants (from reference) ----------------
#define B_DIM   16
#define C_DIM   512
#define HW      3136          // 56*56
#define HW4     784           // HW / 4 (float4 chunks per channel slab)
#define THRESH  8.0f
#define T_CAP   16            // data gives T==2; cap for ws safety
#define REP_CAP (C_DIM * T_CAP)

#ifndef __has_builtin
#define __has_builtin(x) 0
#endif

#if __has_builtin(__builtin_amdgcn_tensor_load_to_lds)
#define HAVE_TDM 1
#if defined(__AMDGCN__)
#if __clang_major__ >= 23
#warning "CDNA5 probe: TDM path ENABLED (6-arg tensor_load_to_lds, clang>=23)"
#else
#warning "CDNA5 probe: TDM path ENABLED (5-arg tensor_load_to_lds, clang<23)"
#endif
#endif
#else
#define HAVE_TDM 0
#if defined(__AMDGCN__)
#warning "CDNA5 probe: TDM builtin MISSING - using direct-gather fallback"
#endif
#endif

// ---------------- TDM helpers (gfx1250 Tensor Data Mover) ----------------
#if HAVE_TDM
typedef __attribute__((ext_vector_type(4))) unsigned int v4u;
typedef __attribute__((ext_vector_type(8))) int          v8i;
typedef __attribute__((ext_vector_type(4))) int          v4i;

// Issue a 1-D TDM tile load: nelem f32 from global -> LDS byte offset.
// Descriptor bit layout per CDNA5 ISA ch.8 (D# groups 0/1; groups 2/3 unused,
// tile_dim1..4 = 0 => 1-D tile).
__device__ __forceinline__ void tdm_load_to_lds(const float* gsrc,
                                                uint32_t lds_byte_off,
                                                int nelem) {
    uint64_t ga = (uint64_t)(uintptr_t)gsrc;
    v4u g0;
    g0.x = 1u;                                           // count=1 (valid), user mode, no gather
    g0.y = lds_byte_off;                                 // lds_addr [63:32]
    g0.z = (uint32_t)ga;                                 // global_addr[31:0]
    g0.w = (uint32_t)((ga >> 32) & 0x01FFFFFFull)        // global_addr[56:32]
         | (2u << 30);                                   // type=2 ("image")
    v8i g1;
    g1[0] = (int)(2u << 16);                             // workgroup_mask=0, data_size=2 (4 bytes)
    g1[1] = (int)(((uint32_t)nelem & 0xFFFFu) << 16);    // tensor_dim0[15:0] in bits 63:48
    g1[2] = (int)(((uint32_t)nelem >> 16) & 0xFFFFu);    // tensor_dim0[31:16]; tensor_dim1=0
    g1[3] = (int)(((uint32_t)nelem & 0xFFFFu) << 16);    // tile_dim0 in bits 127:112
    g1[4] = 0;                                           // tile_dim1=0, tile_dim2=0
    g1[5] = nelem;                                       // tensor_dim0_stride[31:0]
    g1[6] = 0;
    g1[7] = 0;
    v4i gz = {0, 0, 0, 0};
#if __clang_major__ >= 23
    v8i gz8 = {0, 0, 0, 0, 0, 0, 0, 0};
    __builtin_amdgcn_tensor_load_to_lds(g0, g1, gz, gz, gz8, 0);
#else
    __builtin_amdgcn_tensor_load_to_lds(g0, g1, gz, gz, 0);
#endif
}

__device__ __forceinline__ uint32_t lds_offset(const void* p) {
    // flat LDS address: high 32 = aperture, low 32 = offset within WG LDS
    return (uint32_t)(uintptr_t)p;
}
#endif // HAVE_TDM

template <int N>
__device__ __forceinline__ void wait_tensorcnt() {
#if __has_builtin(__builtin_amdgcn_s_wait_tensorcnt)
    __builtin_amdgcn_s_wait_tensorcnt(N);
#else
    asm volatile("s_wait_tensorcnt %0" :: "i"(N) : "memory");
#endif
}

// ---------------- kernel 3 (placed FIRST so disasm snippet shows TDM ops):
// gather + scale. Persistent blocks over (out-channel j, batch b) pairs; each
// pair is one contiguous 12.5 KB slab. Wave 0 issues TDM DMA global->LDS,
// double-buffered via s_wait_tensorcnt, then all threads do scaled b128 stores.
__global__ void cd_gather_kernel(const float* __restrict__ x,
                                 float* __restrict__ out,
                                 const int* __restrict__ hdr,
                                 const int* __restrict__ rep_src,
                                 const float* __restrict__ rep_scale) {
    const int C_out = hdr[0];
    const long total = (long)C_out * B_DIM;
    const int t = threadIdx.x;
#if HAVE_TDM
    __shared__ float lbuf[2][HW];                 // 2 x 12544 B double buffer
    const bool issuer = (t >> 5) == 0;            // wave 0 only (uniform branch)
    long p0 = (long)blockIdx.x;
    if (issuer && p0 < total) {
        int j = (int)(p0 >> 4);
        int b = (int)(p0 & 15);
        const float* src = x + ((size_t)b * C_DIM + (size_t)rep_src[j]) * HW;
        tdm_load_to_lds(src, lds_offset(&lbuf[0][0]), HW);
    }
    int it = 0;
    for (long p = p0; p < total; p += gridDim.x, ++it) {
        long pn = p + gridDim.x;
        if (issuer) {
            if (pn < total) {
                int jn = (int)(pn >> 4);
                int bn = (int)(pn & 15);
                const float* srcn =
                    x + ((size_t)bn * C_DIM + (size_t)rep_src[jn]) * HW;
                tdm_load_to_lds(srcn, lds_offset(&lbuf[(it + 1) & 1][0]), HW);
                wait_tensorcnt<1>();              // oldest (current buffer) done
            } else {
                wait_tensorcnt<0>();
            }
        }
        __syncthreads();                          // LDS tile visible to all waves
        int j = (int)(p >> 4);
        int b = (int)(p & 15);
        float s = rep_scale[j];
        const float4* __restrict__ l4 = (const float4*)&lbuf[it & 1][0];
        float4* __restrict__ dst4 =
            (float4*)(out + ((size_t)b * (size_t)C_out + (size_t)j) * HW);
        for (int v = t; v < HW4; v += 256) {
            float4 d = l4[v];
            d.x *= s; d.y *= s; d.z *= s; d.w *= s;
            dst4[v] = d;
        }
        __syncthreads();                          // done reading before reuse
    }
#else
    // fallback: direct global gather (also what the host pass parses)
    for (long p = (long)blockIdx.x; p < total; p += gridDim.x) {
        int j = (int)(p >> 4);
        int b = (int)(p & 15);
        float s = rep_scale[j];
        const float4* __restrict__ src4 =
            (const float4*)(x + ((size_t)b * C_DIM + (size_t)rep_src[j]) * HW);
        float4* __restrict__ dst4 =
            (float4*)(out + ((size_t)b * (size_t)C_out + (size_t)j) * HW);
        for (int v = t; v < HW4; v += 256) {
            float4 d = src4[v];
            d.x *= s; d.y *= s; d.z *= s; d.w *= s;
            dst4[v] = d;
        }
    }
#endif
}

// ---------------- kernel 1: per-channel max|x| ----------------
// one block per channel; float4 streaming loads (b-outer loop keeps addressing
// as pure strided adds, no integer division); LDS max-tree.
__global__ void cd_chmax_kernel(const float* __restrict__ x,
                                float* __restrict__ chmax) {
    const int c = blockIdx.x;
    const int t = threadIdx.x;
    const float4* __restrict__ x4 = (const float4*)x;
    float m = 0.0f;
    for (int b = 0; b < B_DIM; ++b) {
        const float4* __restrict__ base =
            x4 + ((size_t)b * C_DIM + (size_t)c) * HW4;
        for (int v = t; v < HW4; v += 256) {
            float4 d = base[v];
            m = fmaxf(m, fmaxf(fmaxf(fabsf(d.x), fabsf(d.y)),
                               fmaxf(fabsf(d.z), fabsf(d.w))));
        }
    }
    __shared__ float red[256];
    red[t] = m;
    __syncthreads();
    for (int s = 128; s > 0; s >>= 1) {
        if (t < s) red[t] = fmaxf(red[t], red[t + s]);
        __syncthreads();
    }
    if (t == 0) chmax[c] = red[0];
}

// ---------------- kernel 2: plan (scan) ----------------
// single block, 512 threads (one per channel). Builds rep_src / rep_scale,
// header {C_out, N_out}, and writes the outlier-index tail of d_out.
__global__ void cd_plan_kernel(const float* __restrict__ chmax,
                               int* __restrict__ hdr,
                               int* __restrict__ rep_src,
                               float* __restrict__ rep_scale,
                               float* __restrict__ out) {
    const int t = threadIdx.x;            // channel id
    float m = chmax[t];
    int flag = (m > THRESH) ? 1 : 0;
    int T = 0;
    if (flag) {
        T = (int)ceilf(m / THRESH);
        if (T < 1) T = 1;
        if (T > T_CAP) T = T_CAP;
    }
    __shared__ int sT[C_DIM];
    __shared__ int sF[C_DIM];
    sT[t] = T;
    sF[t] = flag;
    __syncthreads();
    // Hillis-Steele inclusive scan over 512 entries
    for (int off = 1; off < C_DIM; off <<= 1) {
        int aT = (t >= off) ? sT[t - off] : 0;
        int aF = (t >= off) ? sF[t - off] : 0;
        __syncthreads();
        sT[t] += aT;
        sF[t] += aF;
        __syncthreads();
    }
    const int C_out  = sT[C_DIM - 1];
    const int startT = sT[t] - T;
    const float invT = flag ? (1.0f / (float)T) : 0.0f;
    for (int k = 0; k < T; ++k) {
        rep_src[startT + k]   = t;
        rep_scale[startT + k] = invT;
    }
    if (flag) {
        size_t tail = (size_t)B_DIM * (size_t)C_out * (size_t)HW;
        out[tail + (size_t)(sF[t] - 1)] = (float)t;   // outlier_idx, in order
    }
    if (t == 0) {
        hdr[0] = C_out;
        hdr[1] = sF[C_DIM - 1];
    }
}

// ---------------- launch ----------------
// ws layout (bytes):
//   [0,    2048) : chmax[512] float
//   [2048, 2056) : hdr {C_out, N_out}
//   [2112, +32K) : rep_src[REP_CAP] int
//   [+32K, +64K) : rep_scale[REP_CAP] float
extern "C" void kernel_launch(void* const* d_in, const int* in_sizes, int n_in,
                              void* d_out, int out_size, void* d_ws, size_t ws_size,
                              hipStream_t stream) {
    (void)in_sizes; (void)n_in; (void)out_size; (void)ws_size;
    const float* x = (const float*)d_in[0];
    float* out = (float*)d_out;
    char* ws = (char*)d_ws;
    float* chmax     = (float*)ws;
    int*   hdr       = (int*)(ws + 2048);
    int*   rep_src   = (int*)(ws + 2112);
    float* rep_scale = (float*)(ws + 2112 + 4 * REP_CAP);

    cd_chmax_kernel<<<C_DIM, 256, 0, stream>>>(x, chmax);
    cd_plan_kernel<<<1, C_DIM, 0, stream>>>(chmax, hdr, rep_src, rep_scale, out);
    cd_gather_kernel<<<2048, 256, 0, stream>>>(x, out, hdr, rep_src, rep_scale);
}